// IRGLoss_34256659153337
// MI455X (gfx1250) — compile-verified
//
#include <hip/hip_runtime.h>
#include <hip/hip_bf16.h>

typedef float v2f __attribute__((ext_vector_type(2)));
typedef float v8f __attribute__((ext_vector_type(8)));

#define N_ROWS 4096
#define D_FEAT 256
#define N_CLS  1000
#define WF     0.8f
#define N_TILES_UT 1056   // #(bi,bj) tiles with bj >= 2*bi ; bi<32, bj<64

__device__ __forceinline__ float waveReduceSum(float v) {
    #pragma unroll
    for (int off = 16; off > 0; off >>= 1)
        v += __shfl_xor(v, off, 32);
    return v;
}

__device__ __forceinline__ void blockReduceAtomic(float v, float* acc, float scale) {
    __shared__ float smem[8];
    const int lane = threadIdx.x & 31;
    const int wave = threadIdx.x >> 5;
    v = waveReduceSum(v);
    if (lane == 0) smem[wave] = v;
    __syncthreads();
    if (wave == 0) {
        float s = (lane < (int)(blockDim.x >> 5)) ? smem[lane] : 0.0f;
        s = waveReduceSum(s);
        if (lane == 0) atomicAdd(acc, s * scale);
    }
}

__global__ void init_acc_kernel(float* ws) {
    if (threadIdx.x < 16) ws[threadIdx.x] = 0.0f;
}

// One wave per row: sum of squares of one 256-float row.
__global__ __launch_bounds__(256) void rowsqn_kernel(const float* __restrict__ f,
                                                     float* __restrict__ out) {
    const int wave = threadIdx.x >> 5;
    const int lane = threadIdx.x & 31;
    const int row  = blockIdx.x * (blockDim.x >> 5) + wave;
    if (row >= N_ROWS) return;
    const float* p = f + (size_t)row * D_FEAT;
    float s = 0.0f;
    #pragma unroll
    for (int k = lane; k < D_FEAT; k += 32) {
        float x = p[k];
        s = fmaf(x, x, s);
    }
    s = waveReduceSum(s);
    if (lane == 0) out[row] = s;
}

__global__ __launch_bounds__(256) void labelmse_kernel(const float* __restrict__ p,
                                                       const float* __restrict__ t,
                                                       float* acc, int n) {
    float s = 0.0f;
    for (int i = blockIdx.x * blockDim.x + threadIdx.x; i < n; i += gridDim.x * blockDim.x) {
        float d = p[i] - t[i];
        s = fmaf(d, d, s);
    }
    blockReduceAtomic(s, acc, (1.0f - WF) / (float)n);
}

// Fused dual-Gram (pred + target) + pairwise-L2 + squared-diff reduction,
// exploiting symmetry: only upper-triangular 128x64 tiles are computed and
// the result is doubled (diagonal contributes ~0 since d[i,i]==0).
// Block = 8 wave32s covering a 128(i) x 64(j) tile; each wave does 32x32.
// fp32 WMMA 16x16x4: A/B fragments are v2f, C/D are v8f.
__global__ __launch_bounds__(256) void gram_dist_kernel(
    const float* __restrict__ pf0, const float* __restrict__ pf1, const float* __restrict__ pf2,
    const float* __restrict__ tf0, const float* __restrict__ tf1, const float* __restrict__ tf2,
    const float* __restrict__ sqn,   // [6][N_ROWS]: pred 0..2 then target 0..2
    float* __restrict__ acc)
{
    const int fid = blockIdx.y;
    const float* PF = (fid == 0) ? pf0 : (fid == 1) ? pf1 : pf2;
    const float* TF = (fid == 0) ? tf0 : (fid == 1) ? tf1 : tf2;
    const float* sqnP = sqn + (size_t)fid * N_ROWS;
    const float* sqnT = sqn + (size_t)(fid + 3) * N_ROWS;

    // Map linear tile index -> (bi, bj) with bj >= 2*bi (upper + straddle tiles).
    // Per bi there are 64 - 2*bi tiles. Small uniform SALU loop (<=32 iters).
    int t = blockIdx.x;
    int bi = 0;
    while (t >= 64 - 2 * bi) { t -= 64 - 2 * bi; ++bi; }
    const int bj = 2 * bi + t;
    const bool straddle = (bj < 2 * bi + 2);  // tile touches/crosses the diagonal

    const int lane = threadIdx.x & 31;
    const int wave = threadIdx.x >> 5;
    const int wi = wave & 3;   // 0..3 -> i sub-block of 32 within 128
    const int wj = wave >> 2;  // 0..1 -> j sub-block of 32 within 64
    const int i0 = bi * 128 + wi * 32;
    const int j0 = bj * 64  + wj * 32;

    const int lr = lane & 15;  // fragment row (A) / col (B)
    const int kh = lane >> 4;  // which half of K-slab: +2

    // Per-lane base offsets in floats (fit easily in 32 bits: 4096*256 = 1M).
    const unsigned rA0 = (unsigned)(i0      + lr) * D_FEAT + 2u * kh;
    const unsigned rA1 = (unsigned)(i0 + 16 + lr) * D_FEAT + 2u * kh;
    const unsigned rB0 = (unsigned)(j0      + lr) * D_FEAT + 2u * kh;
    const unsigned rB1 = (unsigned)(j0 + 16 + lr) * D_FEAT + 2u * kh;

    v8f aP00 = {}, aP01 = {}, aP10 = {}, aP11 = {};
    v8f aT00 = {}, aT01 = {}, aT10 = {}, aT11 = {};

    #pragma unroll 2
    for (int k0 = 0; k0 < D_FEAT; k0 += 4) {
        v2f Ap0 = *(const v2f*)(PF + rA0 + k0);
        v2f Ap1 = *(const v2f*)(PF + rA1 + k0);
        v2f Bp0 = *(const v2f*)(PF + rB0 + k0);
        v2f Bp1 = *(const v2f*)(PF + rB1 + k0);
        v2f At0 = *(const v2f*)(TF + rA0 + k0);
        v2f At1 = *(const v2f*)(TF + rA1 + k0);
        v2f Bt0 = *(const v2f*)(TF + rB0 + k0);
        v2f Bt1 = *(const v2f*)(TF + rB1 + k0);

        aP00 = __builtin_amdgcn_wmma_f32_16x16x4_f32(false, Ap0, false, Bp0, (short)0, aP00, false, false);
        aP01 = __builtin_amdgcn_wmma_f32_16x16x4_f32(false, Ap0, false, Bp1, (short)0, aP01, false, false);
        aP10 = __builtin_amdgcn_wmma_f32_16x16x4_f32(false, Ap1, false, Bp0, (short)0, aP10, false, false);
        aP11 = __builtin_amdgcn_wmma_f32_16x16x4_f32(false, Ap1, false, Bp1, (short)0, aP11, false, false);
        aT00 = __builtin_amdgcn_wmma_f32_16x16x4_f32(false, At0, false, Bt0, (short)0, aT00, false, false);
        aT01 = __builtin_amdgcn_wmma_f32_16x16x4_f32(false, At0, false, Bt1, (short)0, aT01, false, false);
        aT10 = __builtin_amdgcn_wmma_f32_16x16x4_f32(false, At1, false, Bt0, (short)0, aT10, false, false);
        aT11 = __builtin_amdgcn_wmma_f32_16x16x4_f32(false, At1, false, Bt1, (short)0, aT11, false, false);
    }

    // Epilogue: distances + squared-difference, fully in registers.
    // C layout: VGPR v, lanes 0-15 -> row v, col lane; lanes 16-31 -> row v+8, col lane-16.
    float lsum = 0.0f;
    #pragma unroll
    for (int r = 0; r < 2; ++r) {
        #pragma unroll
        for (int c = 0; c < 2; ++c) {
            const v8f gp = (r == 0) ? (c == 0 ? aP00 : aP01) : (c == 0 ? aP10 : aP11);
            const v8f gt = (r == 0) ? (c == 0 ? aT00 : aT01) : (c == 0 ? aT10 : aT11);
            const int gj = j0 + 16 * c + lr;
            const float sqPj = sqnP[gj];
            const float sqTj = sqnT[gj];
            const int giBase = i0 + 16 * r + 8 * kh;
            #pragma unroll
            for (int v = 0; v < 8; ++v) {
                const int gi = giBase + v;
                float sp = sqnP[gi] + sqPj - 2.0f * gp[v];
                float st = sqnT[gi] + sqTj - 2.0f * gt[v];
                float dp = (sp > 0.0f) ? sqrtf(sp) : 0.0f;
                float dt = (st > 0.0f) ? sqrtf(st) : 0.0f;
                float d  = dp - dt;
                // Straddle tiles keep only strictly-upper elements (diag d==0).
                float wgt = (!straddle || gj > gi) ? 1.0f : 0.0f;
                lsum = fmaf(d * d, wgt, lsum);
            }
        }
    }
    // x2 for symmetry (lower triangle mirrored).
    blockReduceAtomic(lsum, acc, 2.0f);
}

__global__ void finalize_kernel(const float* __restrict__ ws, float* __restrict__ out) {
    if (threadIdx.x == 0) {
        const float label = ws[0];  // already scaled by (1-WF)/(N*C)
        const float feat  = ws[1];  // sum over 3 features of sum((dp-dt)^2) over NxN
        out[0] = label + (WF / 3.0f) * (feat / ((float)N_ROWS * (float)N_ROWS));
    }
}

extern "C" void kernel_launch(void* const* d_in, const int* in_sizes, int n_in,
                              void* d_out, int out_size, void* d_ws, size_t ws_size,
                              hipStream_t stream) {
    (void)in_sizes; (void)n_in; (void)out_size; (void)ws_size;

    const float* pred   = (const float*)d_in[0];
    const float* target = (const float*)d_in[1];
    const float* pf[3]  = { (const float*)d_in[2], (const float*)d_in[3], (const float*)d_in[4] };
    const float* tf[3]  = { (const float*)d_in[5], (const float*)d_in[6], (const float*)d_in[7] };

    float* ws  = (float*)d_ws;       // ws[0]=label acc, ws[1]=feature acc
    float* sqn = ws + 16;            // [6][N_ROWS] row squared norms
    float* out = (float*)d_out;

    init_acc_kernel<<<1, 32, 0, stream>>>(ws);

    // 6 matrices x 4096 rows; 8 rows (waves) per 256-thread block.
    for (int f = 0; f < 3; ++f) {
        rowsqn_kernel<<<N_ROWS / 8, 256, 0, stream>>>(pf[f], sqn + (size_t)f * N_ROWS);
        rowsqn_kernel<<<N_ROWS / 8, 256, 0, stream>>>(tf[f], sqn + (size_t)(3 + f) * N_ROWS);
    }

    labelmse_kernel<<<1024, 256, 0, stream>>>(pred, target, ws + 0, N_ROWS * N_CLS);

    dim3 grid(N_TILES_UT, 3);
    gram_dist_kernel<<<grid, 256, 0, stream>>>(pf[0], pf[1], pf[2],
                                               tf[0], tf[1], tf[2],
                                               sqn, ws + 1);

    finalize_kernel<<<1, 32, 0, stream>>>(ws, out);
}